// TVNet_7413113553400
// MI455X (gfx1250) — compile-verified
//
#include <hip/hip_runtime.h>
#include <stdint.h>
#include <math.h>

// ---------------- problem constants (reference: B=8, C=1, H=W=480) ----------
#define HH 480
#define WW 480
#define TAU_C   0.25f
#define LBDA_C  0.15f
#define THETA_C 0.3f
#define LT_C    (LBDA_C * THETA_C)   // l_t = lambda*theta
#define TAUT_C  (TAU_C / THETA_C)    // tau/theta
#define EPS_C   1e-12f

// ---------------- tiling: 32x8 threads = 256 = 8 waves (wave32) -------------
#define TX 32
#define TY 8
#define TW (TX + 1)
#define THT (TY + 1)
#define TILE (TW * THT)

__device__ __forceinline__ int clampi(int v, int lo, int hi) {
    return v < lo ? lo : (v > hi ? hi : v);
}

// low 32 bits of a generic pointer to a __shared__ object == LDS byte offset
__device__ __forceinline__ uint32_t lds_off(const void* p) {
    return (uint32_t)(uintptr_t)p;
}

// CDNA5 async global->LDS copy (tracked by ASYNCcnt). VDST = LDS byte addr.
__device__ __forceinline__ void async_g2lds_b32(uint32_t ldsoff, const float* g) {
    asm volatile("global_load_async_to_lds_b32 %0, %1, off"
                 :
                 : "v"(ldsoff), "v"((unsigned long long)(uintptr_t)g)
                 : "memory");
}

__device__ __forceinline__ void wait_async0() {
#if __has_builtin(__builtin_amdgcn_s_wait_asynccnt)
    __builtin_amdgcn_s_wait_asynccnt(0);
#else
    asm volatile("s_wait_asynccnt 0" ::: "memory");
#endif
}

// ---------------- centered gradient of x2 (one-sided at borders) ------------
__global__ void k_cgrad(const float* __restrict__ x2,
                        float* __restrict__ d2x, float* __restrict__ d2y) {
    int x = blockIdx.x * TX + threadIdx.x;
    int y = blockIdx.y * TY + threadIdx.y;
    int b = blockIdx.z;
    if (x >= WW || y >= HH) return;
    size_t base = (size_t)b * (HH * WW);
    const float* im = x2 + base;
    int xm = clampi(x - 1, 0, WW - 1), xp = clampi(x + 1, 0, WW - 1);
    int ym = clampi(y - 1, 0, HH - 1), yp = clampi(y + 1, 0, HH - 1);
    size_t i = base + (size_t)y * WW + x;
    d2x[i] = 0.5f * (im[(size_t)y * WW + xp] - im[(size_t)y * WW + xm]);
    d2y[i] = 0.5f * (im[(size_t)yp * WW + x] - im[(size_t)ym * WW + x]);
}

// ---------------- bilinear warp of x2, d2x, d2y by flow (u1,u2) -------------
// reproduces STN grid math: g = linspace(-1,1,N); coord = (g + u*2/N + 1)*N/2
__global__ void k_warp(const float* __restrict__ x1, const float* __restrict__ x2,
                       const float* __restrict__ d2x, const float* __restrict__ d2y,
                       const float* __restrict__ u1, const float* __restrict__ u2,
                       float* __restrict__ dxw, float* __restrict__ dyw,
                       float* __restrict__ grad, float* __restrict__ rho_c) {
    int x = blockIdx.x * TX + threadIdx.x;
    int y = blockIdx.y * TY + threadIdx.y;
    int b = blockIdx.z;
    if (x >= WW || y >= HH) return;
    size_t base = (size_t)b * (HH * WW);
    size_t i = base + (size_t)y * WW + x;

    float uu = u1[i], vv = u2[i];
    float gx = -1.0f + 2.0f * (float)x / (float)(WW - 1);
    float gy = -1.0f + 2.0f * (float)y / (float)(HH - 1);
    float xm = (gx + uu * (2.0f / (float)WW) + 1.0f) * ((float)WW * 0.5f);
    float ym = (gy + vv * (2.0f / (float)HH) + 1.0f) * ((float)HH * 0.5f);

    int x0 = (int)floorf(xm);
    int y0 = (int)floorf(ym);
    int x0i = clampi(x0, 0, WW - 1), x1i = clampi(x0 + 1, 0, WW - 1);
    int y0i = clampi(y0, 0, HH - 1), y1i = clampi(y0 + 1, 0, HH - 1);
    float x0f = (float)x0i, x1f = (float)x1i;
    float y0f = (float)y0i, y1f = (float)y1i;
    float wa = (x1f - xm) * (y1f - ym);
    float wb = (x1f - xm) * (ym - y0f);
    float wc = (xm - x0f) * (y1f - ym);
    float wd = (xm - x0f) * (ym - y0f);

    size_t ia = base + (size_t)y0i * WW + x0i;
    size_t ib = base + (size_t)y1i * WW + x0i;
    size_t ic = base + (size_t)y0i * WW + x1i;
    size_t id = base + (size_t)y1i * WW + x1i;

    float x2w = wa * x2[ia] + wb * x2[ib] + wc * x2[ic] + wd * x2[id];
    float dx  = wa * d2x[ia] + wb * d2x[ib] + wc * d2x[ic] + wd * d2x[id];
    float dy  = wa * d2y[ia] + wb * d2y[ib] + wc * d2y[ic] + wd * d2y[id];

    float g  = dx * dx + dy * dy + EPS_C;
    float rc = x2w - dx * uu - dy * vv - x1[i];
    dxw[i] = dx; dyw[i] = dy; grad[i] = g; rho_c[i] = rc;
}

// ---------------- u-update: thresholding + THETA*div(p) (u in place) --------
// div needs p at (x,y),(x-1,y),(x,y-1): stage p tiles with top/left halo in LDS
// via CDNA5 async-to-LDS, then read neighborhoods with ds_load.
__global__ void k_update_u(float* __restrict__ u1, float* __restrict__ u2,
                           float* __restrict__ rho_out,
                           const float* __restrict__ p11, const float* __restrict__ p12,
                           const float* __restrict__ p21, const float* __restrict__ p22,
                           const float* __restrict__ dxw, const float* __restrict__ dyw,
                           const float* __restrict__ grad, const float* __restrict__ rho_c) {
    __shared__ float sp11[TILE], sp12[TILE], sp21[TILE], sp22[TILE];
    int tx = threadIdx.x, ty = threadIdx.y;
    int tid = ty * TX + tx;
    int bx = blockIdx.x * TX, by = blockIdx.y * TY, b = blockIdx.z;
    size_t base = (size_t)b * (HH * WW);

    for (int idx = tid; idx < TILE; idx += TX * TY) {
        int lx = idx % TW, ly = idx / TW;
        int gx = clampi(bx - 1 + lx, 0, WW - 1);
        int gy = clampi(by - 1 + ly, 0, HH - 1);
        size_t gi = base + (size_t)gy * WW + gx;
        async_g2lds_b32(lds_off(&sp11[idx]), &p11[gi]);
        async_g2lds_b32(lds_off(&sp12[idx]), &p12[gi]);
        async_g2lds_b32(lds_off(&sp21[idx]), &p21[gi]);
        async_g2lds_b32(lds_off(&sp22[idx]), &p22[gi]);
    }
    wait_async0();
    __syncthreads();

    int x = bx + tx, y = by + ty;
    if (x >= WW || y >= HH) return;
    size_t i = base + (size_t)y * WW + x;
    int c = (ty + 1) * TW + (tx + 1);

    // backward-difference divergence with reference boundary rules
    float d1 = (x == 0) ? sp11[c] : ((x == WW - 1) ? -sp11[c - 1] : sp11[c] - sp11[c - 1]);
    d1      += (y == 0) ? sp12[c] : ((y == HH - 1) ? -sp12[c - TW] : sp12[c] - sp12[c - TW]);
    float d2 = (x == 0) ? sp21[c] : ((x == WW - 1) ? -sp21[c - 1] : sp21[c] - sp21[c - 1]);
    d2      += (y == 0) ? sp22[c] : ((y == HH - 1) ? -sp22[c - TW] : sp22[c] - sp22[c - TW]);

    float uu = u1[i], vv = u2[i];
    float dx = dxw[i], dy = dyw[i], g = grad[i];
    float rho = rho_c[i] + dx * uu + dy * vv + EPS_C;
    float ltg = LT_C * g;
    float s;
    if (rho < -ltg)      s = LT_C;
    else if (rho > ltg)  s = -LT_C;
    else if (g > EPS_C)  s = -rho / g;
    else                 s = 0.0f;

    u1[i] = s * dx + uu + THETA_C * d1;
    u2[i] = s * dy + vv + THETA_C * d2;
    rho_out[i] = rho;   // final inner iteration of final warp leaves reference rho
}

// ---------------- p-update: forward gradients of u (p in place) -------------
__global__ void k_update_p(const float* __restrict__ u1, const float* __restrict__ u2,
                           float* __restrict__ p11, float* __restrict__ p12,
                           float* __restrict__ p21, float* __restrict__ p22) {
    __shared__ float su1[TILE], su2[TILE];
    int tx = threadIdx.x, ty = threadIdx.y;
    int tid = ty * TX + tx;
    int bx = blockIdx.x * TX, by = blockIdx.y * TY, b = blockIdx.z;
    size_t base = (size_t)b * (HH * WW);

    for (int idx = tid; idx < TILE; idx += TX * TY) {
        int lx = idx % TW, ly = idx / TW;
        int gx = clampi(bx + lx, 0, WW - 1);
        int gy = clampi(by + ly, 0, HH - 1);
        size_t gi = base + (size_t)gy * WW + gx;
        async_g2lds_b32(lds_off(&su1[idx]), &u1[gi]);
        async_g2lds_b32(lds_off(&su2[idx]), &u2[gi]);
    }
    wait_async0();
    __syncthreads();

    int x = bx + tx, y = by + ty;
    if (x >= WW || y >= HH) return;
    size_t i = base + (size_t)y * WW + x;
    int c = ty * TW + tx;

    float a = su1[c];
    float u1x = (x < WW - 1) ? su1[c + 1]  - a : 0.0f;
    float u1y = (y < HH - 1) ? su1[c + TW] - a : 0.0f;
    float bq = su2[c];
    float u2x = (x < WW - 1) ? su2[c + 1]  - bq : 0.0f;
    float u2y = (y < HH - 1) ? su2[c + TW] - bq : 0.0f;

    float n1 = 1.0f + TAUT_C * sqrtf(u1x * u1x + u1y * u1y + EPS_C);
    float n2 = 1.0f + TAUT_C * sqrtf(u2x * u2x + u2y * u2y + EPS_C);
    float r1 = 1.0f / n1, r2 = 1.0f / n2;

    p11[i] = (p11[i] + TAUT_C * u1x) * r1;
    p12[i] = (p12[i] + TAUT_C * u1y) * r1;
    p21[i] = (p21[i] + TAUT_C * u2x) * r2;
    p22[i] = (p22[i] + TAUT_C * u2y) * r2;
}

// ---------------- host launcher ---------------------------------------------
extern "C" void kernel_launch(void* const* d_in, const int* in_sizes, int n_in,
                              void* d_out, int out_size, void* d_ws, size_t ws_size,
                              hipStream_t stream) {
    const float* x1   = (const float*)d_in[0];
    const float* x2   = (const float*)d_in[1];
    const float* u1in = (const float*)d_in[2];
    const float* u2in = (const float*)d_in[3];

    const size_t N = (size_t)in_sizes[0];        // B*1*H*W
    const int B = (int)(N / ((size_t)HH * WW));

    float* out = (float*)d_out;
    float* u1  = out;                // evolving flow lives directly in d_out
    float* u2  = out + N;
    float* rho = out + 2 * N;

    float* ws  = (float*)d_ws;
    float* d2x = ws + 0 * N;
    float* d2y = ws + 1 * N;
    float* p11 = ws + 2 * N;         // p11,p12,p21,p22 contiguous for one memset
    float* p12 = ws + 3 * N;
    float* p21 = ws + 4 * N;
    float* p22 = ws + 5 * N;
    float* dxw = ws + 6 * N;
    float* dyw = ws + 7 * N;
    float* grd = ws + 8 * N;
    float* rhc = ws + 9 * N;

    hipMemcpyAsync(u1, u1in, N * sizeof(float), hipMemcpyDeviceToDevice, stream);
    hipMemcpyAsync(u2, u2in, N * sizeof(float), hipMemcpyDeviceToDevice, stream);
    hipMemsetAsync(p11, 0, 4 * N * sizeof(float), stream);   // p := 0 every call

    dim3 blk(TX, TY, 1);
    dim3 grid((WW + TX - 1) / TX, (HH + TY - 1) / TY, B);

    k_cgrad<<<grid, blk, 0, stream>>>(x2, d2x, d2y);
    for (int w = 0; w < 5; ++w) {
        k_warp<<<grid, blk, 0, stream>>>(x1, x2, d2x, d2y, u1, u2, dxw, dyw, grd, rhc);
        for (int it = 0; it < 5; ++it) {
            k_update_u<<<grid, blk, 0, stream>>>(u1, u2, rho, p11, p12, p21, p22,
                                                 dxw, dyw, grd, rhc);
            k_update_p<<<grid, blk, 0, stream>>>(u1, u2, p11, p12, p21, p22);
        }
    }
}